// motion_mask_loss_60275571032829
// MI455X (gfx1250) — compile-verified
//
#include <hip/hip_runtime.h>
#include <stdint.h>

// ---------------------------------------------------------------- constants
#define HH 512
#define WW 512
#define NBOX 160                   // 5*32 boxes per batch
#define WPR 16                     // 512 cols / 32 bits = 16 mask words per row
#define MASK_WORDS (HH * WPR)      // 8192 words (32 KiB) per batch
#define BPB 16                     // main-kernel blocks per batch
#define NSTAGE 16                  // float4-stages per wave (4096 f4 / 8 waves / 32 lanes)

// workspace layout (bytes)
#define OFF_ACC  0                 // float acc[2*B]  (S_all, S_in)
#define OFF_N    1024              // int   ncount[B]
#define OFF_MASK 4096              // uint32 mask[B * MASK_WORDS]

typedef float v4f __attribute__((ext_vector_type(4)));  // native clang vector

#if defined(__gfx1250__) && \
    __has_builtin(__builtin_amdgcn_global_load_async_to_lds_b128) && \
    __has_builtin(__builtin_amdgcn_s_wait_asynccnt)
#define USE_ASYNC 1
#else
#define USE_ASYNC 0
#endif

#if USE_ASYNC
typedef int v4i __attribute__((ext_vector_type(4)));
typedef __attribute__((address_space(1))) v4i* as1_v4i_p;   // ptr to global int4
typedef __attribute__((address_space(3))) v4i* as3_v4i_p;   // ptr to LDS int4
__device__ __forceinline__ void async_ld_b128(const void* g, void* l) {
  __builtin_amdgcn_global_load_async_to_lds_b128((as1_v4i_p)g, (as3_v4i_p)l, 0, 0);
}
#endif

__device__ __forceinline__ void accum4(const v4f v, uint32_t bits,
                                       float& s_all, float& s_in) {
  const float p0 = 1.0f / (1.0f + __expf(-v.x));
  const float p1 = 1.0f / (1.0f + __expf(-v.y));
  const float p2 = 1.0f / (1.0f + __expf(-v.z));
  const float p3 = 1.0f / (1.0f + __expf(-v.w));
  s_all += (p0 + p1) + (p2 + p3);
  s_in  += ((bits & 1u) ? p0 : 0.0f) + ((bits & 2u) ? p1 : 0.0f) +
           ((bits & 4u) ? p2 : 0.0f) + ((bits & 8u) ? p3 : 0.0f);
}

// ---------------------------------------------------------------- kernel 0
__global__ __launch_bounds__(256) void zero_acc_kernel(float* acc, int n) {
  const int i = blockIdx.x * 256 + threadIdx.x;
  if (i < n) acc[i] = 0.0f;
}

// ---------------------------------------------------------------- kernel 1
// One block per batch: paint 160 boxes into a 512x512 bitmask held in LDS,
// popcount it (-> N), and write it to global workspace.
__global__ __launch_bounds__(256) void build_mask_kernel(
    const int* __restrict__ boxes,    // [B, NBOX, 4] = x, y, sx, sy
    uint32_t* __restrict__ maskg,     // [B, MASK_WORDS]
    int* __restrict__ ncount) {       // [B]
  __shared__ uint32_t smask[MASK_WORDS];           // 32 KiB
  __shared__ int      bx1[NBOX], bx2[NBOX];
  __shared__ uint32_t bwlo[NBOX], bwhi[NBOX], bmlo[NBOX], bmhi[NBOX];
  __shared__ int      wred[8];

  const int b   = blockIdx.x;
  const int tid = threadIdx.x;

  for (int i = tid; i < MASK_WORDS; i += 256) smask[i] = 0u;

  if (tid < NBOX) {
    const int* g = boxes + (size_t)(b * NBOX + tid) * 4;
    const int x = g[0], y = g[1], sx = g[2], sy = g[3];
    int x1 = min(x, HH), x2 = min(x + sx, HH);
    int y1 = min(y, WW), y2 = min(y + sy, WW);
    if (x2 <= x1 || y2 <= y1) { x1 = 1; x2 = 0; y1 = 0; y2 = 1; }  // empty box
    bx1[tid] = x1; bx2[tid] = x2;
    bwlo[tid] = (uint32_t)y1 >> 5;
    bwhi[tid] = (uint32_t)(y2 - 1) >> 5;
    bmlo[tid] = 0xFFFFFFFFu << ((uint32_t)y1 & 31u);
    bmhi[tid] = 0xFFFFFFFFu >> (31u - ((uint32_t)(y2 - 1) & 31u));
  }
  __syncthreads();

  // one (box,row) pair per trip; consecutive lanes = consecutive rows (coherent)
  for (int i = tid; i < NBOX * HH; i += 256) {
    const int box = i >> 9;          // i / 512
    const int row = i & (HH - 1);    // i % 512
    if (row >= bx1[box] && row < bx2[box]) {
      const uint32_t wlo = bwlo[box], whi = bwhi[box];
      const uint32_t mlo = bmlo[box], mhi = bmhi[box];
      for (uint32_t w = wlo; w <= whi; ++w) {
        uint32_t v = 0xFFFFFFFFu;
        if (w == wlo) v &= mlo;
        if (w == whi) v &= mhi;
        atomicOr(&smask[row * WPR + w], v);
      }
    }
  }
  __syncthreads();

  // popcount + 128-bit writeback
  int cnt = 0;
  uint4*       mg4 = (uint4*)(maskg + (size_t)b * MASK_WORDS);
  const uint4* sm4 = (const uint4*)smask;
  for (int i = tid; i < MASK_WORDS / 4; i += 256) {
    const uint4 v = sm4[i];
    mg4[i] = v;
    cnt += __popc(v.x) + __popc(v.y) + __popc(v.z) + __popc(v.w);
  }
  for (int off = 16; off; off >>= 1) cnt += __shfl_xor(cnt, off, 32);
  if ((tid & 31) == 0) wred[tid >> 5] = cnt;
  __syncthreads();
  if (tid == 0) {
    int t = 0;
    for (int i = 0; i < 8; ++i) t += wred[i];
    ncount[b] = t;
  }
}

// ---------------------------------------------------------------- kernel 2
// Streaming masked sigmoid reduction. Each wave owns 16 private LDS slots;
// all 16 global_load_async_to_lds_b128 stages are issued up front (8 KiB/wave
// in flight), then drained with per-stage constant s_wait_asynccnt(15-s).
__global__ __launch_bounds__(256) void dice_main_kernel(
    const float* __restrict__ pred,
    const uint32_t* __restrict__ maskg,
    float* __restrict__ acc) {
  const int b     = blockIdx.x / BPB;
  const int chunk = blockIdx.x % BPB;
  const int wave  = threadIdx.x >> 5;
  const int lane  = threadIdx.x & 31;

  const int qbase = chunk * 4096 + wave * 512;        // float4 index in image
  const v4f*      p4 = (const v4f*)pred + (size_t)b * (HH * WW / 4);
  const uint32_t* bm = maskg + (size_t)b * MASK_WORDS;

  float s_all = 0.0f, s_in = 0.0f;

#if USE_ASYNC
  __shared__ v4f sbuf[8 * NSTAGE * 32];               // 64 KiB: 8 waves x 16 slots x 512B
  v4f* mybuf = sbuf + wave * (NSTAGE * 32) + lane;    // wave-private, lane-strided

#pragma unroll
  for (int s = 0; s < NSTAGE; ++s)                    // issue everything up front
    async_ld_b128(p4 + qbase + s * 32 + lane, mybuf + s * 32);

  // drain: stage s is complete once ASYNCcnt <= 15-s (in-order completion)
#define STAGE(s)                                                          \
  do {                                                                    \
    const int q   = qbase + (s) * 32 + lane;                              \
    const int row = q >> 7;                                               \
    const int c4  = q & 127;                                              \
    const uint32_t mw   = bm[row * WPR + (c4 >> 3)];                      \
    const uint32_t bits = (mw >> ((c4 & 7) * 4)) & 0xFu;                  \
    __builtin_amdgcn_s_wait_asynccnt(NSTAGE - 1 - (s));                   \
    asm volatile("" ::: "memory");                                        \
    const v4f v = mybuf[(s) * 32];                                        \
    accum4(v, bits, s_all, s_in);                                         \
  } while (0)

  STAGE(0);  STAGE(1);  STAGE(2);  STAGE(3);
  STAGE(4);  STAGE(5);  STAGE(6);  STAGE(7);
  STAGE(8);  STAGE(9);  STAGE(10); STAGE(11);
  STAGE(12); STAGE(13); STAGE(14); STAGE(15);
#undef STAGE
#else
  for (int s = 0; s < NSTAGE; ++s) {
    const int q   = qbase + s * 32 + lane;
    const int row = q >> 7;
    const int c4  = q & 127;
    const uint32_t mw   = bm[row * WPR + (c4 >> 3)];
    const uint32_t bits = (mw >> ((c4 & 7) * 4)) & 0xFu;
    const v4f v = __builtin_nontemporal_load(&p4[q]);
    accum4(v, bits, s_all, s_in);
  }
#endif

  // wave32 butterfly reduction, then 2 f32 atomics per wave
  for (int off = 16; off; off >>= 1) {
    s_all += __shfl_xor(s_all, off, 32);
    s_in  += __shfl_xor(s_in,  off, 32);
  }
  if (lane == 0) {
    atomicAdd(&acc[2 * b],     s_all);
    atomicAdd(&acc[2 * b + 1], s_in);
  }
}

// ---------------------------------------------------------------- kernel 3
__global__ __launch_bounds__(32) void finalize_kernel(
    const float* __restrict__ acc, const int* __restrict__ ncount,
    float* __restrict__ out, int B) {
  const int t = threadIdx.x;
  float sum = 0.0f;
  for (int b = t; b < B; b += 32) {
    const float s_all = acc[2 * b];
    const float inter = 255.0f * acc[2 * b + 1];
    const float n255  = 255.0f * (float)ncount[b];
    sum += (inter + 1.0f) / (s_all + n255 - inter + 1.0f);
  }
  for (int off = 16; off; off >>= 1) sum += __shfl_xor(sum, off, 32);
  if (t == 0) out[0] = 1.0f - sum / (float)B;
}

// ---------------------------------------------------------------- launcher
extern "C" void kernel_launch(void* const* d_in, const int* in_sizes, int n_in,
                              void* d_out, int out_size, void* d_ws, size_t ws_size,
                              hipStream_t stream) {
  const float* pred  = (const float*)d_in[0];
  const int*   boxes = (const int*)d_in[1];
  const int B = in_sizes[1] / (NBOX * 4);   // 64

  float*    acc    = (float*)((char*)d_ws + OFF_ACC);
  int*      ncount = (int*)((char*)d_ws + OFF_N);
  uint32_t* maskg  = (uint32_t*)((char*)d_ws + OFF_MASK);

  zero_acc_kernel<<<(2 * B + 255) / 256, 256, 0, stream>>>(acc, 2 * B);
  build_mask_kernel<<<B, 256, 0, stream>>>(boxes, maskg, ncount);
  dice_main_kernel<<<B * BPB, 256, 0, stream>>>(pred, maskg, acc);
  finalize_kernel<<<1, 32, 0, stream>>>(acc, ncount, (float*)d_out, B);
}